// MultiHeadDensityRatioEstimator_6691559047273
// MI455X (gfx1250) — compile-verified
//
#include <hip/hip_runtime.h>
#include <hip/hip_bf16.h>

typedef float v2f __attribute__((ext_vector_type(2)));
typedef float v8f __attribute__((ext_vector_type(8)));

#define LN2f     0.6931471805599453f
#define INVLN2f  1.4426950408889634f

static __device__ __forceinline__ float fast_log(float x) {   // natural log
  return __builtin_amdgcn_logf(x) * LN2f;                     // v_log_f32 is log2
}
static __device__ __forceinline__ float fast_exp(float x) {   // natural exp
  return __builtin_amdgcn_exp2f(x * INVLN2f);
}
static __device__ __forceinline__ float fast_rcp(float x) {
  return __builtin_amdgcn_rcpf(x);
}

constexpr int N  = 4096;
constexpr int Dm = 128;
constexpr int H  = 8;
constexpr int KA = 20;   // augmented K: 16 data + [norm,1] + pad to multiple of 4

// workspace layout (float offsets); total ~5.4 MB (assumes ws_size >= 6 MB)
constexpr size_t WS_YAUG = 0;                                 // [H][N][KA]
constexpr size_t WS_XAUG = WS_YAUG + (size_t)H * N * KA;      // [H][N][KA]
constexpr size_t WS_R    = WS_XAUG + (size_t)H * N * KA;      // [H][N] row recip sums
constexpr size_t WS_LP   = WS_R + (size_t)H * N;              // [N] diag sum_h log1p(sq)
constexpr size_t WS_SC   = WS_LP + N;                         // scalars
// sc[0]=T1 (sum offdiag L), sc[1]=decaySum, sc[2]=LpSum, sc[3]=BL,
// sc[4]=tp, sc[5]=sigNegSum, sc[6]=fp

__global__ void zero_kernel(float* sc) {
  if (threadIdx.x < 32) sc[threadIdx.x] = 0.0f;
}

// Build augmented matrices so that  y'.x' = ||y||^2 + ||x||^2 - 2 y.x = sq.
// Also zero R and accumulate sum of squares for the decay term.
__global__ __launch_bounds__(256) void prep_kernel(
    const float* __restrict__ zx, const float* __restrict__ zy,
    float* __restrict__ Yaug, float* __restrict__ Xaug,
    float* __restrict__ R, float* __restrict__ sc) {
  int t = blockIdx.x * blockDim.x + threadIdx.x;   // t = h*N + i
  int i = t & (N - 1);
  int h = t >> 12;
  const float* y = zy + (size_t)i * Dm + h * 16;
  const float* x = zx + (size_t)i * Dm + h * 16;
  float yn = 0.0f, xn = 0.0f;
  float yv[16], xv[16];
#pragma unroll
  for (int k = 0; k < 16; ++k) {
    yv[k] = y[k]; yn += yv[k] * yv[k];
    xv[k] = x[k]; xn += xv[k] * xv[k];
  }
  float* yo = Yaug + (size_t)t * KA;
  float* xo = Xaug + (size_t)t * KA;
#pragma unroll
  for (int k = 0; k < 16; ++k) { yo[k] = -2.0f * yv[k]; xo[k] = xv[k]; }
  yo[16] = yn;   yo[17] = 1.0f; yo[18] = 0.0f; yo[19] = 0.0f;
  xo[16] = 1.0f; xo[17] = xn;   xo[18] = 0.0f; xo[19] = 0.0f;
  R[t] = 0.0f;
  float d = yn + xn;
  for (int o = 16; o > 0; o >>= 1) d += __shfl_down(d, o, 32);
  if ((threadIdx.x & 31) == 0) atomicAdd(&sc[1], d);
}

// Diagonal: Lp_i = sum_h log1p(||zy_i - zx_i||^2_h)
__global__ __launch_bounds__(256) void diag_kernel(
    const float* __restrict__ zx, const float* __restrict__ zy,
    float* __restrict__ Lp, float* __restrict__ sc) {
  int i = blockIdx.x * blockDim.x + threadIdx.x;
  float L = 0.0f;
#pragma unroll
  for (int h = 0; h < H; ++h) {
    float sq = 0.0f;
#pragma unroll
    for (int k = 0; k < 16; ++k) {
      float d = zy[(size_t)i * Dm + h * 16 + k] - zx[(size_t)i * Dm + h * 16 + k];
      sq += d * d;
    }
    L += fast_log(1.0f + sq);
  }
  Lp[i] = L;
  float s = L;
  for (int o = 16; o > 0; o >>= 1) s += __shfl_down(s, o, 32);
  if ((threadIdx.x & 31) == 0) atomicAdd(&sc[2], s);
}

// Main pair pass. One wave owns a 16-row strip x 256 columns (16 col tiles).
// PASS 1: accumulate R_h(i) (off-diag recip row sums) and T1 = sum offdiag L.
// PASS 2: accumulate sum sigmoid(avg-neg) and fp count (needs BL = sc[3]).
template <int PASS>
__global__ __launch_bounds__(256) void pair_kernel(
    const float* __restrict__ Yaug, const float* __restrict__ Xaug,
    float* __restrict__ R, float* __restrict__ sc) {
  const int lid      = threadIdx.x & 31;
  const int wave     = (blockIdx.x * blockDim.x + threadIdx.x) >> 5;  // 0..4095
  const int rowTile  = wave >> 4;        // 0..255
  const int colChunk = wave & 15;        // 0..15
  const int row_base = rowTile * 16;
  const int m16 = lid & 15;              // A: M index, B: N index, C: N index
  const int hb  = lid >> 4;              // lane half selects K pair / M half

  // A fragments (f32 16x4 layout): lane holds (M=m16, K=slab*4 + hb*2 + {0,1})
  v2f a[H][5];
#pragma unroll
  for (int h = 0; h < H; ++h) {
    const float* yr = Yaug + ((size_t)h * N + row_base + m16) * KA + hb * 2;
#pragma unroll
    for (int s = 0; s < 5; ++s) a[h][s] = *(const v2f*)(yr + s * 4);
  }

  float racc[H][8];
  if (PASS == 1) {
#pragma unroll
    for (int h = 0; h < H; ++h)
#pragma unroll
      for (int v = 0; v < 8; ++v) racc[h][v] = 0.0f;
  }
  float Lacc = 0.0f, sigacc = 0.0f, fpacc = 0.0f;
  const float BL = (PASS == 2) ? sc[3] : 0.0f;

  for (int t = 0; t < 16; ++t) {
    const int col_base = (colChunk * 16 + t) * 16;
    v8f c[H];
#pragma unroll
    for (int h = 0; h < H; ++h) {
      const float* xr = Xaug + ((size_t)h * N + col_base + m16) * KA + hb * 2;
      v8f acc = {};
#pragma unroll
      for (int s = 0; s < 5; ++s) {
        v2f b = *(const v2f*)(xr + s * 4);
        // D = A x B + C  (16x16x4 f32); accumulator IS sq directly.
        acc = __builtin_amdgcn_wmma_f32_16x16x4_f32(
            false, a[h][s], false, b, (short)0, acc, false, false);
      }
      c[h] = acc;
    }
    // Epilogue: C layout lane=(N=m16, M=v+8*hb).
#pragma unroll
    for (int v = 0; v < 8; ++v) {
      const int row = row_base + v + hb * 8;
      const int col = col_base + m16;
      const bool off = (row != col);
      float prod = 1.0f;
#pragma unroll
      for (int h = 0; h < H; ++h) {
        float sq = fmaxf(c[h][v], 0.0f);
        float op = 1.0f + sq;
        prod *= op;                    // prod = PI_h (1+sq_h)
        if (PASS == 1) racc[h][v] += off ? fast_rcp(op) : 0.0f;  // exp(logit)
      }
      const float L = fast_log(prod);  // = sum_h log1p(sq_h)
      if (PASS == 1) {
        Lacc += off ? L : 0.0f;
      } else {
        // avg-neg = (-L - BL)/8 ; sigmoid(avg-neg) = 1/(1+exp((L+BL)/8))
        const float s = fast_rcp(1.0f + fast_exp((L + BL) * 0.125f));
        sigacc += off ? s : 0.0f;
        fpacc  += (off && (L < -BL)) ? 1.0f : 0.0f;   // sigmoid > 0.5
      }
    }
  }

  if (PASS == 1) {
    // reduce within each 16-lane half (rows v and v+8), then global atomic
#pragma unroll
    for (int h = 0; h < H; ++h)
#pragma unroll
      for (int v = 0; v < 8; ++v) {
        float x = racc[h][v];
        x += __shfl_xor(x, 1, 32);
        x += __shfl_xor(x, 2, 32);
        x += __shfl_xor(x, 4, 32);
        x += __shfl_xor(x, 8, 32);
        if (m16 == 0) atomicAdd(&R[h * N + row_base + v + hb * 8], x);
      }
    float x = Lacc;
    for (int o = 16; o > 0; o >>= 1) x += __shfl_down(x, o, 32);
    if (lid == 0) atomicAdd(&sc[0], x);
  } else {
    float x = sigacc, y = fpacc;
    for (int o = 16; o > 0; o >>= 1) {
      x += __shfl_down(x, o, 32);
      y += __shfl_down(y, o, 32);
    }
    if (lid == 0) { atomicAdd(&sc[5], x); atomicAdd(&sc[6], y); }
  }
}

// Single-block: baseline, bl, loss, pos-side stats. Writes out[0,1,2,5,7,8].
__global__ __launch_bounds__(256) void mid_kernel(
    const float* __restrict__ R, const float* __restrict__ Lp,
    float* __restrict__ sc, float* __restrict__ out) {
  __shared__ float sS[H];
  __shared__ float sLR, sSig, sTp;
  const int tid = threadIdx.x;
  if (tid < H) sS[tid] = 0.0f;
  if (tid == 8) sLR = 0.0f;
  if (tid == 9) { sSig = 0.0f; sTp = 0.0f; }
  __syncthreads();

  float ls[H];
#pragma unroll
  for (int h = 0; h < H; ++h) ls[h] = 0.0f;
  float lr = 0.0f;
  for (int i = tid; i < N; i += 256) {
#pragma unroll
    for (int h = 0; h < H; ++h) {
      float r = R[h * N + i];
      ls[h] += r;
      lr += fast_log(r);               // sum_h sum_i log R_h(i)
    }
  }
#pragma unroll
  for (int h = 0; h < H; ++h) atomicAdd(&sS[h], ls[h]);
  atomicAdd(&sLR, lr);
  __syncthreads();

  const float Nf = (float)N;
  const float Nneg = Nf * (Nf - 1.0f);
  float BL = 0.0f;                     // sum_h bl_h (uniform across threads)
#pragma unroll
  for (int h = 0; h < H; ++h) BL += fast_log(sS[h]) - fast_log(Nneg);

  float sig = 0.0f, tp = 0.0f;
  for (int i = tid; i < N; i += 256) {
    float lp = Lp[i];                  // avg-pos_i = (-lp - BL)/8
    sig += fast_rcp(1.0f + fast_exp((lp + BL) * 0.125f));
    tp  += (lp < -BL) ? 1.0f : 0.0f;
  }
  atomicAdd(&sSig, sig);
  atomicAdd(&sTp, tp);
  __syncthreads();

  if (tid == 0) {
    const float T1 = sc[0], dsum = sc[1], LpSum = sc[2];
    const float decay   = 0.01f * dsum / (Nf * (float)Dm);
    const float meanPos = (-LpSum / Nf - BL) * 0.125f;
    const float meanNeg = (-T1 / Nneg - BL) * 0.125f;
    const float meanAtt = -meanPos;
    const float meanRep = (sLR / Nf - 8.0f * fast_log(Nf - 1.0f) - BL) * 0.125f;
    out[0] = meanPos;
    out[1] = meanNeg;
    out[2] = sSig / Nf;                // mean sigmoid(pos)
    out[5] = sTp / Nf;                 // recall
    out[7] = BL * 0.125f;              // bl
    out[8] = meanAtt + meanRep + decay;
    sc[3] = BL;
    sc[4] = sTp;
  }
}

__global__ void final_kernel(const float* __restrict__ sc, float* __restrict__ out) {
  if (blockIdx.x == 0 && threadIdx.x == 0) {
    const float Nf = 4096.0f, Nneg = 4096.0f * 4095.0f;
    const float tp = sc[4], sigNeg = sc[5], fp = sc[6];
    out[3] = sigNeg / Nneg;                       // mean sigmoid(neg)
    const float tn = Nneg - fp;
    out[4] = (tp + tn) / (Nf + Nneg);             // accuracy (denom = 2^24)
    out[6] = (tp + fp > 0.0f) ? tp / fmaxf(tp + fp, 1.0f) : 0.0f;  // precision
  }
}

extern "C" void kernel_launch(void* const* d_in, const int* in_sizes, int n_in,
                              void* d_out, int out_size, void* d_ws, size_t ws_size,
                              hipStream_t stream) {
  const float* zx = (const float*)d_in[0];
  const float* zy = (const float*)d_in[1];
  float* out = (float*)d_out;
  float* w   = (float*)d_ws;

  float* Yaug = w + WS_YAUG;
  float* Xaug = w + WS_XAUG;
  float* R    = w + WS_R;
  float* Lp   = w + WS_LP;
  float* sc   = w + WS_SC;

  zero_kernel<<<1, 64, 0, stream>>>(sc);
  prep_kernel<<<(H * N) / 256, 256, 0, stream>>>(zx, zy, Yaug, Xaug, R, sc);
  diag_kernel<<<N / 256, 256, 0, stream>>>(zx, zy, Lp, sc);
  pair_kernel<1><<<512, 256, 0, stream>>>(Yaug, Xaug, R, sc);   // 4096 waves
  mid_kernel<<<1, 256, 0, stream>>>(R, Lp, sc, out);
  pair_kernel<2><<<512, 256, 0, stream>>>(Yaug, Xaug, R, sc);
  final_kernel<<<1, 32, 0, stream>>>(sc, out);
}